// GNNRankTask0_35381940584594
// MI455X (gfx1250) — compile-verified
//
#include <hip/hip_runtime.h>
#include <math.h>

#define HID 64

typedef __attribute__((ext_vector_type(2))) float v2f;
typedef __attribute__((ext_vector_type(8))) float v8f;

static inline int cdiv_ll(long long a, int b) { return (int)((a + b - 1) / b); }

// ---------------- utility ----------------
__global__ void k_zero(float* __restrict__ p, long long n) {
  long long i = (long long)blockIdx.x * blockDim.x + threadIdx.x;
  if (i < n) p[i] = 0.0f;
}

// ---------------- degrees / norms ----------------
__global__ void k_degree(const int* __restrict__ src, const int* __restrict__ dst,
                         float* __restrict__ deg_out, float* __restrict__ deg_in, int E) {
  int e = blockIdx.x * blockDim.x + threadIdx.x;
  if (e < E) {
    atomicAdd(deg_out + src[e], 1.0f);
    atomicAdd(deg_in + dst[e], 1.0f);
  }
}

__global__ void k_norm(float* __restrict__ deg, int n) {
  int i = blockIdx.x * blockDim.x + threadIdx.x;
  if (i < n) {
    float d = deg[i];
    d = d < 1.0f ? 1.0f : d;           // clip(deg, 1.0)
    deg[i] = 1.0f / sqrtf(d);          // deg^-0.5
  }
}

// ---------------- embedding lookup ----------------
__global__ void k_embed(const int* __restrict__ feat, const float* __restrict__ emb,
                        float* __restrict__ h, int N) {
  long long i = (long long)blockIdx.x * blockDim.x + threadIdx.x;
  if (i < (long long)N * HID) {
    int node = (int)(i >> 6);
    int c = (int)(i & 63);
    h[i] = emb[(long long)feat[node] * HID + c];
  }
}

// ---------------- edge gather + scatter-add (L2-resident atomics) ----------------
// 16 threads per edge, each handling a 16B (float4) chunk -> coalesced 256B rows.
__global__ void k_scatter(const float* __restrict__ h, const float* __restrict__ norm_out,
                          const int* __restrict__ src, const int* __restrict__ dst,
                          float* __restrict__ agg, int E) {
  long long gid = (long long)blockIdx.x * blockDim.x + threadIdx.x;
  if (gid >= (long long)E * 16) return;
  int e = (int)(gid >> 4);
  int c = (int)(gid & 15) << 2;
  int s = src[e];
  int d = dst[e];
  float no = norm_out[s];
  const float4 hv = *(const float4*)(h + (long long)s * HID + c);
  float* ap = agg + (long long)d * HID + c;
  atomicAdd(ap + 0, hv.x * no);
  atomicAdd(ap + 1, hv.y * no);
  atomicAdd(ap + 2, hv.z * no);
  atomicAdd(ap + 3, hv.w * no);
}

// ---------------- dense layer: h = relu((agg * norm_in) @ W + b) via f32 WMMA ----------------
// One wave per 16-row node tile. V_WMMA_F32_16X16X4_F32:
//   A 16x4 f32: lanes 0-15 hold rows M=0..15 with K={0,1} in v0/v1; lanes 16-31 same rows K={2,3}
//   B 4x16 f32: lanes 0-15 hold cols N=0..15 with K={0,1}; lanes 16-31 same cols K={2,3}
//   C/D 16x16 f32: VGPR v, lane L -> row v + 8*(L>=16), col L%16
__global__ __launch_bounds__(32) void k_gemm_relu(
    const float* __restrict__ agg, const float* __restrict__ norm_in,
    const float* __restrict__ W, const float* __restrict__ bias,
    float* __restrict__ hout, int N) {
  const int lane  = threadIdx.x;
  const int mrow  = lane & 15;
  const int khalf = lane >> 4;           // 0 -> K{0,1}, 1 -> K{2,3} within each 4-chunk
  const int tile0 = blockIdx.x * 16;
  const int row   = tile0 + mrow;
  const int rcl   = row < N ? row : N - 1;
  float scale = norm_in[rcl];
  if (row >= N) scale = 0.0f;            // zero-pad tail rows (keeps EXEC uniform for WMMA)
  const float* arow = agg + (long long)rcl * HID;

  // Preload all 16 A fragments (K = 0..63 in chunks of 4), scaled by norm_in.
  v2f afrag[16];
#pragma unroll
  for (int kc = 0; kc < 16; ++kc) {
    float2 av = *(const float2*)(arow + kc * 4 + khalf * 2);
    afrag[kc].x = av.x * scale;
    afrag[kc].y = av.y * scale;
  }

#pragma unroll
  for (int nt = 0; nt < 4; ++nt) {       // four 16-col output tiles
    const int n0 = nt * 16;
    v8f c = {};
#pragma unroll
    for (int kc = 0; kc < 16; ++kc) {    // K = 64 in steps of 4
      const int k0 = kc * 4 + khalf * 2;
      v2f bfrag;
      bfrag.x = W[(k0 + 0) * HID + n0 + mrow];
      bfrag.y = W[(k0 + 1) * HID + n0 + mrow];
      c = __builtin_amdgcn_wmma_f32_16x16x4_f32(
          /*neg_a=*/false, afrag[kc], /*neg_b=*/false, bfrag,
          /*c_mod=*/(short)0, c, /*reuse_a=*/false, /*reuse_b=*/false);
    }
    const float bv = bias[n0 + mrow];
#pragma unroll
    for (int v = 0; v < 8; ++v) {
      int orow = tile0 + v + 8 * khalf;
      if (orow < N) {
        float val = c[v] + bv;
        hout[(long long)orow * HID + n0 + mrow] = val > 0.0f ? val : 0.0f;
      }
    }
  }
}

// ---------------- per-graph mean pooling ----------------
__global__ void k_pool_sum(const float* __restrict__ h, const int* __restrict__ gids,
                           float* __restrict__ gsums, int N) {
  long long i = (long long)blockIdx.x * blockDim.x + threadIdx.x;
  if (i < (long long)N * HID) {
    int node = (int)(i >> 6);
    int c = (int)(i & 63);
    atomicAdd(gsums + (long long)gids[node] * HID + c, h[i]);
  }
}

__global__ void k_pool_cnt(const int* __restrict__ gids, float* __restrict__ gcnt, int N) {
  int i = blockIdx.x * blockDim.x + threadIdx.x;
  if (i < N) atomicAdd(gcnt + gids[i], 1.0f);
}

__global__ void k_finalize(const float* __restrict__ gsums, const float* __restrict__ gcnt,
                           float* __restrict__ out, int total) {
  int i = blockIdx.x * blockDim.x + threadIdx.x;
  if (i < total) {
    float cv = gcnt[i >> 6];
    cv = cv < 1.0f ? 1.0f : cv;
    out[i] = gsums[i] / cv;
  }
}

// ---------------- driver ----------------
extern "C" void kernel_launch(void* const* d_in, const int* in_sizes, int n_in,
                              void* d_out, int out_size, void* d_ws, size_t ws_size,
                              hipStream_t stream) {
  // setup_inputs() dict order:
  const int*   node_feat = (const int*)d_in[0];
  const int*   src       = (const int*)d_in[1];
  const int*   dst       = (const int*)d_in[2];
  const int*   graph_ids = (const int*)d_in[3];
  const float* emb       = (const float*)d_in[4];
  const float* Ws[3]     = {(const float*)d_in[5], (const float*)d_in[7], (const float*)d_in[9]};
  const float* bs[3]     = {(const float*)d_in[6], (const float*)d_in[8], (const float*)d_in[10]};
  float* out = (float*)d_out;

  const int N = in_sizes[0];
  const int E = in_sizes[1];
  const int G = out_size / HID;

  // Workspace layout (floats)
  float* ws       = (float*)d_ws;
  float* deg_out  = ws;                                   // N   (becomes norm_out in place)
  float* deg_in   = ws + (long long)N;                    // N   (becomes norm_in in place)
  float* h        = ws + 2LL * N;                         // N*64
  float* agg      = h + (long long)N * HID;               // N*64
  float* gsums    = agg + (long long)N * HID;             // G*64
  float* gcnt     = gsums + (long long)G * HID;           // G

  const int B256 = 256;

  // 1) zero degrees + pooling accumulators (fresh every call)
  k_zero<<<cdiv_ll(2LL * N, B256), B256, 0, stream>>>(deg_out, 2LL * N);
  k_zero<<<cdiv_ll((long long)G * HID + G, B256), B256, 0, stream>>>(gsums, (long long)G * HID + G);

  // 2) degrees -> norms
  k_degree<<<cdiv_ll(E, B256), B256, 0, stream>>>(src, dst, deg_out, deg_in, E);
  k_norm<<<cdiv_ll(2LL * N, B256), B256, 0, stream>>>(deg_out, 2 * N);  // both arrays contiguous

  // 3) embedding lookup
  k_embed<<<cdiv_ll((long long)N * HID, B256), B256, 0, stream>>>(node_feat, emb, h, N);

  // 4) three GCN layers
  const int gemm_blocks = (N + 15) / 16;
  for (int l = 0; l < 3; ++l) {
    k_zero<<<cdiv_ll((long long)N * HID, B256), B256, 0, stream>>>(agg, (long long)N * HID);
    k_scatter<<<cdiv_ll((long long)E * 16, B256), B256, 0, stream>>>(h, deg_out, src, dst, agg, E);
    k_gemm_relu<<<gemm_blocks, 32, 0, stream>>>(agg, deg_in, Ws[l], bs[l], h, N);
  }

  // 5) per-graph mean pooling
  k_pool_sum<<<cdiv_ll((long long)N * HID, B256), B256, 0, stream>>>(h, graph_ids, gsums, N);
  k_pool_cnt<<<cdiv_ll(N, B256), B256, 0, stream>>>(graph_ids, gcnt, N);
  k_finalize<<<cdiv_ll(out_size, B256), B256, 0, stream>>>(gsums, gcnt, out, out_size);
}